// AtomCenteredTensorMomentDescriptor_50148038148220
// MI455X (gfx1250) — compile-verified
//
#include <hip/hip_runtime.h>
#include <math.h>

#define L_MAX    4
#define NUM_LM   25
#define R_BASIS  16
#define F_DIM    64
#define FE_DIM   144
#define CUTOFF_R 5.0f

typedef float v2f __attribute__((ext_vector_type(2)));
typedef float v8f __attribute__((ext_vector_type(8)));

struct CgEntry { unsigned ab; float coeff; };           // a | (b<<16), global lm indices
struct CgGroup { int c; int s; int path; int start; int count; };

// ---------------- workspace layout (bytes) ----------------
static const size_t OFF_HDR     = 0;                    // 2 ints
static const size_t OFF_CG      = 256;                  // 15625 floats
static const size_t OFF_GROUPS  = 64ull * 1024;         // <=1024 CgGroup
static const size_t OFF_ENTRIES = 128ull * 1024;        // <=16384 CgEntry
static const size_t OFF_TE      = 1ull << 20;           // N*144 floats
static const size_t OFF_Y0      = 8ull << 20;           // N*25*16 floats
static const size_t OFF_A       = 32ull << 20;          // N*2*25*64 floats
static const size_t OFF_B       = 160ull << 20;
static const size_t OFF_TD0     = 288ull << 20;
static const size_t OFF_TD1     = 416ull << 20;         // ends ~544 MiB

// =====================================================================
// 1) On-device real Clebsch-Gordan tensor (module constant in the
//    reference, NOT a kernel input -> recompute each launch, fast+deterministic)
// =====================================================================
__device__ __forceinline__ double lfac_d(int n) { return lgamma((double)n + 1.0); }

__device__ double cg_complex_d(int l1,int m1,int l2,int m2,int l3,int m3) {
  if (m1 + m2 != m3 || l3 < abs(l1 - l2) || l3 > l1 + l2) return 0.0;
  double pre = 0.5 * (lfac_d(l1+l2-l3) + lfac_d(l1-l2+l3) + lfac_d(-l1+l2+l3)
                      - lfac_d(l1+l2+l3+1)
                      + lfac_d(l1+m1) + lfac_d(l1-m1) + lfac_d(l2+m2) + lfac_d(l2-m2)
                      + lfac_d(l3+m3) + lfac_d(l3-m3));
  int kmin = max(0, max(l2 - l3 - m1, l1 - l3 + m2));
  int kmax = min(l1 + l2 - l3, min(l1 - m1, l2 + m2));
  double s = 0.0;
  for (int k = kmin; k <= kmax; ++k) {
    double ln = lfac_d(k) + lfac_d(l1+l2-l3-k) + lfac_d(l1-m1-k) + lfac_d(l2+m2-k)
              + lfac_d(l3-l2+m1+k) + lfac_d(l3-l1-m2+k);
    s += ((k & 1) ? -1.0 : 1.0) * exp(pre - ln);
  }
  return sqrt(2.0 * l3 + 1.0) * s;
}

__device__ __forceinline__ int lm_to_l(int i) { int l = 0; while ((l+1)*(l+1) <= i) ++l; return l; }

// rows of the complex->real basis change U (each row has <=2 nonzeros)
__device__ int u_row_d(int i, int* col, double* ur, double* ui) {
  int l = lm_to_l(i); int off = l*l + l; int m = i - off;
  const double inv = 0.70710678118654752440;
  if (m == 0) { col[0] = off; ur[0] = 1.0; ui[0] = 0.0; return 1; }
  if (m > 0) {
    double sg = (m & 1) ? -1.0 : 1.0;
    col[0] = off + m; ur[0] = sg * inv; ui[0] = 0.0;
    col[1] = off - m; ur[1] = inv;      ui[1] = 0.0;
    return 2;
  }
  int mm = -m; double sg = (mm & 1) ? -1.0 : 1.0;
  col[0] = off - mm; ur[0] = 0.0; ui[0] = inv;        // i/sqrt2
  col[1] = off + mm; ur[1] = 0.0; ui[1] = -sg * inv;  // -i*(-1)^m/sqrt2
  return 2;
}

__global__ void cg_dense_kernel(float* __restrict__ cg) {
  int ij = blockIdx.x * blockDim.x + threadIdx.x;
  if (ij >= NUM_LM * NUM_LM) return;
  int i = ij / NUM_LM, j = ij % NUM_LM;
  int ca[2], cb[2], cc[2];
  double ar[2], ai_[2], br[2], bi_[2], cr[2], ci_[2];
  int na = u_row_d(i, ca, ar, ai_);
  int nb = u_row_d(j, cb, br, bi_);
  int li = lm_to_l(i), lj = lm_to_l(j);
  int offi = li*li + li, offj = lj*lj + lj;
  for (int k = 0; k < NUM_LM; ++k) {
    int lk = lm_to_l(k); int offk = lk*lk + lk;
    int nc = u_row_d(k, cc, cr, ci_);
    double Tre = 0.0, Tim = 0.0;
    for (int a = 0; a < na; ++a)
      for (int b = 0; b < nb; ++b) {
        double pr = ar[a]*br[b] - ai_[a]*bi_[b];
        double pi = ar[a]*bi_[b] + ai_[a]*br[b];
        int m1 = ca[a] - offi, m2 = cb[b] - offj;
        for (int cx = 0; cx < nc; ++cx) {
          int m3 = cc[cx] - offk;
          double v = cg_complex_d(li, m1, lj, m2, lk, m3);
          if (v == 0.0) continue;
          // (pr + i*pi) * conj(cr + i*ci) * v
          Tre += (pr * cr[cx] + pi * ci_[cx]) * v;
          Tim += (pi * cr[cx] - pr * ci_[cx]) * v;
        }
      }
    double C = Tre + Tim;
    if (fabs(C) < 1e-12) C = 0.0;
    cg[(i * NUM_LM + j) * NUM_LM + k] = (float)C;
  }
}

// compact dense CG into (group-by-output-c -> entry list), path order matching PATHS
__global__ void build_lists_kernel(const float* __restrict__ cg, CgGroup* __restrict__ groups,
                                   CgEntry* __restrict__ entries, int* __restrict__ hdr) {
  if (threadIdx.x != 0 || blockIdx.x != 0) return;
  int ne = 0, ng = 0, path = 0;
  for (int l1 = 0; l1 <= L_MAX; ++l1)
    for (int l2 = 0; l2 <= L_MAX; ++l2) {
      int lo = abs(l1 - l2), hi2 = min(L_MAX, l1 + l2);
      for (int l3 = lo; l3 <= hi2; ++l3) {
        int s = (l1 + l2 + l3) & 1;
        for (int mc = 0; mc < 2*l3 + 1; ++mc) {
          int c = l3*l3 + mc;
          int start = ne, cnt = 0;
          for (int ma = 0; ma < 2*l1 + 1; ++ma) {
            int a = l1*l1 + ma;
            for (int mb = 0; mb < 2*l2 + 1; ++mb) {
              int b = l2*l2 + mb;
              float cf = cg[(a * NUM_LM + b) * NUM_LM + c];
              if (cf != 0.0f) {
                entries[ne].ab = (unsigned)a | ((unsigned)b << 16);
                entries[ne].coeff = cf;
                ++ne; ++cnt;
              }
            }
          }
          if (cnt) { CgGroup G; G.c = c; G.s = s; G.path = path; G.start = start; G.count = cnt; groups[ng++] = G; }
        }
        ++path;
      }
    }
  hdr[0] = ng; hdr[1] = ne;
}

// =====================================================================
// 2) Edge stage: RBF * cutoff, radial mix, real sph harm, scatter-add
//    thread = (edge, f) with f in [0,16); 25 f32 atomics into L2-resident y0
// =====================================================================
__global__ __launch_bounds__(256) void edge_kernel(
    const int* __restrict__ Zs, const int* __restrict__ nbr,
    const float* __restrict__ disp, const float* __restrict__ Wsp,
    const float* __restrict__ norm, float* __restrict__ y0, int E) {
  int t = blockIdx.x * blockDim.x + threadIdx.x;
  int e = t >> 4, f = t & 15;
  if (e >= E) return;
  int i = nbr[2*e], j = nbr[2*e + 1];
  float dx = disp[3*e], dy = disp[3*e + 1], dz = disp[3*e + 2];
  float r = sqrtf(dx*dx + dy*dy + dz*dz + 1e-12f);
  float inv_r = 1.0f / r;
  float x = dx*inv_r, y = dy*inv_r, z = dz*inv_r;
  float rc = fminf(r * (1.0f / CUTOFF_R), 1.0f);
  float fcut = 0.5f * (cosf(3.14159265358979323846f * rc) + 1.0f);

  int zj = Zs[j];
  const float* w = Wsp + (size_t)zj * (R_BASIS * R_BASIS) + f;
  float g = 0.0f;
  const float gamma = (16.0f / 5.0f) * (16.0f / 5.0f);     // (K/CUTOFF)^2
  #pragma unroll
  for (int k = 0; k < R_BASIS; ++k) {
    float d = r - (float)k * (CUTOFF_R / 15.0f);            // linspace(0,5,16)
    g = fmaf(expf(-gamma * d * d), w[k * R_BASIS], g);
  }
  g *= fcut / norm[0];

  float x2 = x*x, y2 = y*y, z2 = z*z;
  float Y[NUM_LM];
  Y[0]  = 0.28209479177387814f;
  Y[1]  = 0.48860251190291987f * y;
  Y[2]  = 0.48860251190291987f * z;
  Y[3]  = 0.48860251190291987f * x;
  Y[4]  = 1.0925484305920792f * x * y;
  Y[5]  = 1.0925484305920792f * y * z;
  Y[6]  = 0.31539156525252005f * (3.f*z2 - 1.f);
  Y[7]  = 1.0925484305920792f * x * z;
  Y[8]  = 0.5462742152960396f * (x2 - y2);
  Y[9]  = 0.5900435899266435f * y * (3.f*x2 - y2);
  Y[10] = 2.890611442640554f  * x * y * z;
  Y[11] = 0.4570457994644658f * y * (5.f*z2 - 1.f);
  Y[12] = 0.3731763325901154f * z * (5.f*z2 - 3.f);
  Y[13] = 0.4570457994644658f * x * (5.f*z2 - 1.f);
  Y[14] = 1.445305721320277f  * z * (x2 - y2);
  Y[15] = 0.5900435899266435f * x * (x2 - 3.f*y2);
  Y[16] = 2.5033429417967046f * x * y * (x2 - y2);
  Y[17] = 1.7701307697799304f * y * z * (3.f*x2 - y2);
  Y[18] = 0.9461746957575601f * x * y * (7.f*z2 - 1.f);
  Y[19] = 0.6690465435572892f * y * z * (7.f*z2 - 3.f);
  Y[20] = 0.10578554691520431f * (35.f*z2*z2 - 30.f*z2 + 3.f);
  Y[21] = 0.6690465435572892f * x * z * (7.f*z2 - 3.f);
  Y[22] = 0.47308734787878004f * (x2 - y2) * (7.f*z2 - 1.f);
  Y[23] = 1.7701307697799304f * x * z * (x2 - 3.f*y2);
  Y[24] = 0.6258357354491761f * (x2*x2 - 6.f*x2*y2 + y2*y2);

  float* dst = y0 + (size_t)i * (NUM_LM * R_BASIS) + f;
  #pragma unroll
  for (int lm = 0; lm < NUM_LM; ++lm)
    atomicAdd(dst + lm * R_BASIS, Y[lm] * g);
}

// =====================================================================
// 3) degree_dense via V_WMMA_F32_16X16X4_F32 (fp32 matrix cores)
//    Rows = (n,m) for fixed (p,l); N*(2l+1) is a multiple of 16, so tiles
//    never straddle l-regions. One wave computes a 16x64 tile of BOTH
//    a = x@W1[p,l] and b = x@W2[p,l], sharing the A fragments.
// =====================================================================
template<int FIN, int NP>
__global__ __launch_bounds__(32) void dd_kernel(
    const float* __restrict__ x, const float* __restrict__ W1, const float* __restrict__ W2,
    float* __restrict__ aout, float* __restrict__ bout, int N) {
  int l = (int)blockIdx.y % 5;
  int p = (int)blockIdx.y / 5;
  int twol1 = 2*l + 1;
  int rowsL = N * twol1;
  int row0 = (int)blockIdx.x * 16;
  if (row0 >= rowsL) return;                 // uniform per-wave: EXEC stays full for WMMA
  int lane = threadIdx.x;
  int mrow = lane & 15, hi = lane >> 4;
  int r = row0 + mrow;
  int n = r / twol1, m = r - n * twol1;
  int lm = l*l + m;
  const float2* x2p = (const float2*)(x + ((size_t)(n * NP + p) * NUM_LM + lm) * FIN);
  const float* Wl1 = W1 + (size_t)(p * 5 + l) * FIN * F_DIM;
  const float* Wl2 = W2 + (size_t)(p * 5 + l) * FIN * F_DIM;

  v8f z8 = {0.f,0.f,0.f,0.f,0.f,0.f,0.f,0.f};
  v8f acc1[4] = {z8, z8, z8, z8};
  v8f acc2[4] = {z8, z8, z8, z8};

  #pragma unroll
  for (int kk = 0; kk < FIN / 4; ++kk) {
    // A 16x4 f32 fragment: lanes 0-15 rows M, K={0,1}; lanes 16-31 K={2,3}
    float2 at = x2p[kk * 2 + hi];
    v2f av; av.x = at.x; av.y = at.y;
    int k0 = kk * 4 + 2 * hi;
    #pragma unroll
    for (int ct = 0; ct < 4; ++ct) {
      int col = ct * 16 + mrow;
      v2f b1; b1.x = Wl1[(size_t)k0 * F_DIM + col]; b1.y = Wl1[(size_t)(k0+1) * F_DIM + col];
      v2f b2; b2.x = Wl2[(size_t)k0 * F_DIM + col]; b2.y = Wl2[(size_t)(k0+1) * F_DIM + col];
      acc1[ct] = __builtin_amdgcn_wmma_f32_16x16x4_f32(false, av, false, b1, (short)0, acc1[ct], false, false);
      acc2[ct] = __builtin_amdgcn_wmma_f32_16x16x4_f32(false, av, false, b2, (short)0, acc2[ct], false, false);
    }
  }
  // D layout: vgpr v -> M = v + 8*hi, N = lane&15
  #pragma unroll
  for (int ct = 0; ct < 4; ++ct) {
    #pragma unroll
    for (int v = 0; v < 8; ++v) {
      int rr = row0 + v + 8 * hi;
      int nn = rr / twol1, mm = rr - nn * twol1;
      size_t o = ((size_t)(nn * NP + p) * NUM_LM + l*l + mm) * F_DIM + ct * 16 + mrow;
      aout[o] = acc1[ct][v];
      bout[o] = acc2[ct][v];
    }
  }
}

// =====================================================================
// 4) Tensor product: 1 atom/block, 64 threads (thread = feature column).
//    a,b staged as float2 (both parities) in LDS -> one ds_load_b64 per
//    operand per CG entry; accumulators per output-c group live in
//    registers; each thread owns its f column -> no barriers/atomics.
// =====================================================================
template<bool FIRST>
__global__ __launch_bounds__(64) void tp_kernel(
    const float* __restrict__ A, const float* __restrict__ B,
    const float* __restrict__ wp, float* __restrict__ out,
    const int* __restrict__ hdr, const CgGroup* __restrict__ groups,
    const CgEntry* __restrict__ entries) {
  __shared__ float2 sA[NUM_LM * F_DIM];      // 12.8 KB
  __shared__ float2 sB[NUM_LM * F_DIM];      // 12.8 KB
  __shared__ float  sO[2 * NUM_LM * F_DIM];  // 12.8 KB
  int n = blockIdx.x;
  int f = threadIdx.x;

  if (FIRST) {
    const float* Ab = A + (size_t)n * NUM_LM * F_DIM;
    const float* Bb = B + (size_t)n * NUM_LM * F_DIM;
    for (int lm = 0; lm < NUM_LM; ++lm) {
      sA[lm * F_DIM + f] = make_float2(Ab[lm * F_DIM + f], 0.f);
      sB[lm * F_DIM + f] = make_float2(Bb[lm * F_DIM + f], 0.f);
      sO[lm * F_DIM + f] = 0.f;
      sO[(NUM_LM + lm) * F_DIM + f] = 0.f;
    }
  } else {
    const float* Ab = A + (size_t)n * 2 * NUM_LM * F_DIM;
    const float* Bb = B + (size_t)n * 2 * NUM_LM * F_DIM;
    const int ps = NUM_LM * F_DIM;
    for (int lm = 0; lm < NUM_LM; ++lm) {
      sA[lm * F_DIM + f] = make_float2(Ab[lm * F_DIM + f], Ab[ps + lm * F_DIM + f]);
      sB[lm * F_DIM + f] = make_float2(Bb[lm * F_DIM + f], Bb[ps + lm * F_DIM + f]);
      sO[lm * F_DIM + f] = 0.f;
      sO[(NUM_LM + lm) * F_DIM + f] = 0.f;
    }
  }
  // no barrier needed: each thread only touches its own f column

  int nG = hdr[0];
  for (int gi = 0; gi < nG; ++gi) {
    CgGroup G = groups[gi];
    float t00 = 0.f, t01 = 0.f, t10 = 0.f, t11 = 0.f;
    for (int e = G.start; e < G.start + G.count; ++e) {
      CgEntry ent = entries[e];
      int ai = (int)(ent.ab & 0xffffu), bi = (int)(ent.ab >> 16);
      float2 av = sA[ai * F_DIM + f];
      float2 bv = sB[bi * F_DIM + f];
      float cf = ent.coeff;
      t00 = fmaf(cf, av.x * bv.x, t00);
      if (!FIRST) {
        t01 = fmaf(cf, av.x * bv.y, t01);
        t10 = fmaf(cf, av.y * bv.x, t10);
        t11 = fmaf(cf, av.y * bv.y, t11);
      }
    }
    const float* w = wp + (size_t)G.path * 4 * F_DIM + f; // wp[path][p][q][F]
    float even = FIRST ? (w[0] * t00) : fmaf(w[3 * F_DIM], t11, w[0] * t00);
    sO[(G.s * NUM_LM + G.c) * F_DIM + f] += even;
    if (!FIRST) {
      float odd = fmaf(w[2 * F_DIM], t10, w[1 * F_DIM] * t01);
      sO[((1 - G.s) * NUM_LM + G.c) * F_DIM + f] += odd;
    }
  }

  float* ob = out + (size_t)n * 2 * NUM_LM * F_DIM;
  for (int sl = 0; sl < 2 * NUM_LM; ++sl)
    ob[sl * F_DIM + f] = sO[sl * F_DIM + f];
}

// =====================================================================
// 5) te = emb_table[Z] @ W_et + b_et via WMMA f32 (M=N,K=64,Ncols=144)
//    bias folded into the accumulator init (depends only on column N)
// =====================================================================
__global__ __launch_bounds__(32) void te_kernel(
    const int* __restrict__ Z, const float* __restrict__ emb,
    const float* __restrict__ Wet, const float* __restrict__ bet,
    float* __restrict__ te, int N) {
  int lane = threadIdx.x, mrow = lane & 15, hi = lane >> 4;
  int row = (int)blockIdx.x * 16 + mrow;
  int z = Z[row];
  const float2* a2 = (const float2*)(emb + (size_t)z * 64);
  v8f acc[9];
  #pragma unroll
  for (int ct = 0; ct < 9; ++ct) {
    float b = bet[ct * 16 + mrow];
    acc[ct] = (v8f){b, b, b, b, b, b, b, b};
  }
  #pragma unroll
  for (int kk = 0; kk < 16; ++kk) {
    float2 at = a2[kk * 2 + hi];
    v2f av; av.x = at.x; av.y = at.y;
    int k0 = kk * 4 + 2 * hi;
    #pragma unroll
    for (int ct = 0; ct < 9; ++ct) {
      v2f bv;
      bv.x = Wet[(size_t)k0 * FE_DIM + ct * 16 + mrow];
      bv.y = Wet[(size_t)(k0 + 1) * FE_DIM + ct * 16 + mrow];
      acc[ct] = __builtin_amdgcn_wmma_f32_16x16x4_f32(false, av, false, bv, (short)0, acc[ct], false, false);
    }
  }
  #pragma unroll
  for (int ct = 0; ct < 9; ++ct)
    #pragma unroll
    for (int v = 0; v < 8; ++v)
      te[(size_t)((int)blockIdx.x * 16 + v + 8 * hi) * FE_DIM + ct * 16 + mrow] = acc[ct][v];
}

// =====================================================================
// 6) Fused concat + scale + te-add + mish activation, float4 vectorized
// =====================================================================
__constant__ int c_deg[NUM_LM] = {0,1,1,1,2,2,2,2,2,3,3,3,3,3,3,3,4,4,4,4,4,4,4,4,4};

__global__ __launch_bounds__(256) void final_kernel(
    const float* __restrict__ y0, const float* __restrict__ td0,
    const float* __restrict__ td1, const float* __restrict__ te,
    const float* __restrict__ wf, float* __restrict__ out, int N) {
  int idx = blockIdx.x * blockDim.x + threadIdx.x;
  int total = N * 2 * NUM_LM * (FE_DIM / 4);
  if (idx >= total) return;
  int c4 = idx % (FE_DIM / 4); int q = idx / (FE_DIM / 4);
  int lm = q % NUM_LM; q /= NUM_LM;
  int p = q & 1; int n = q >> 1;
  int c = c4 * 4;

  float4 tev = *(const float4*)(te + (size_t)n * FE_DIM + c);
  float4 wfv = *(const float4*)(wf + ((size_t)p * 5 + c_deg[lm]) * FE_DIM + c);
  float4 val;
  if (c < 16) {
    if (p == 0) val = *(const float4*)(y0 + ((size_t)n * NUM_LM + lm) * R_BASIS + c);
    else        val = make_float4(0.f, 0.f, 0.f, 0.f);
  } else if (c < 80) {
    val = *(const float4*)(td0 + (((size_t)n * 2 + p) * NUM_LM + lm) * F_DIM + (c - 16));
  } else {
    val = *(const float4*)(td1 + (((size_t)n * 2 + p) * NUM_LM + lm) * F_DIM + (c - 80));
  }
  float v[4] = { tev.x * val.x * wfv.x, tev.y * val.y * wfv.y,
                 tev.z * val.z * wfv.z, tev.w * val.w * wfv.w };
  if (p == 0 && lm == 0) { v[0] += tev.x; v[1] += tev.y; v[2] += tev.z; v[3] += tev.w; }
  float o[4];
  #pragma unroll
  for (int t = 0; t < 4; ++t) {
    float vv = v[t];
    float sp = (vv > 20.f) ? vv : log1pf(expf(vv));
    o[t] = vv + vv * tanhf(sp);
  }
  *(float4*)(out + (((size_t)n * 2 + p) * NUM_LM + lm) * FE_DIM + c) =
      make_float4(o[0], o[1], o[2], o[3]);
}

// =====================================================================
extern "C" void kernel_launch(void* const* d_in, const int* in_sizes, int n_in,
                              void* d_out, int out_size, void* d_ws, size_t ws_size,
                              hipStream_t stream) {
  (void)n_in; (void)out_size; (void)ws_size;
  const int*   Zs   = (const int*)d_in[0];
  const int*   nbr  = (const int*)d_in[1];
  const float* disp = (const float*)d_in[2];
  const float* Wsp  = (const float*)d_in[3];
  const float* emb  = (const float*)d_in[4];
  const float* Wet  = (const float*)d_in[5];
  const float* bet  = (const float*)d_in[6];
  const float* norm = (const float*)d_in[7];
  const float* W1a  = (const float*)d_in[8];
  const float* W2a  = (const float*)d_in[9];
  const float* wpa  = (const float*)d_in[10];
  const float* W1b  = (const float*)d_in[11];
  const float* W2b  = (const float*)d_in[12];
  const float* wpb  = (const float*)d_in[13];
  const float* wf   = (const float*)d_in[14];
  float* out = (float*)d_out;
  int N = in_sizes[0];
  int E = in_sizes[1] / 2;

  char* ws = (char*)d_ws;
  int*     hdr  = (int*)(ws + OFF_HDR);
  float*   cg   = (float*)(ws + OFF_CG);
  CgGroup* grp  = (CgGroup*)(ws + OFF_GROUPS);
  CgEntry* ent  = (CgEntry*)(ws + OFF_ENTRIES);
  float*   te   = (float*)(ws + OFF_TE);
  float*   y0   = (float*)(ws + OFF_Y0);
  float*   bufA = (float*)(ws + OFF_A);
  float*   bufB = (float*)(ws + OFF_B);
  float*   td0  = (float*)(ws + OFF_TD0);
  float*   td1  = (float*)(ws + OFF_TD1);

  // CG tensor + sparse path lists (recomputed each call; deterministic)
  cg_dense_kernel<<<3, 256, 0, stream>>>(cg);
  build_lists_kernel<<<1, 1, 0, stream>>>(cg, grp, ent, hdr);

  // edge aggregation
  hipMemsetAsync(y0, 0, (size_t)N * NUM_LM * R_BASIS * sizeof(float), stream);
  edge_kernel<<<(E * 16 + 255) / 256, 256, 0, stream>>>(Zs, nbr, disp, Wsp, norm, y0, E);

  // layer 0: degree_dense (K=16, p=0 only) + tensor product (p=0 inputs)
  dim3 g0((unsigned)((N * 9 + 15) / 16), 5);
  dd_kernel<16, 1><<<g0, 32, 0, stream>>>(y0, W1a, W2a, bufA, bufB, N);
  tp_kernel<true><<<N, 64, 0, stream>>>(bufA, bufB, wpa, td0, hdr, grp, ent);

  // layer 1: degree_dense (K=64, both parities) + full tensor product
  dim3 g1((unsigned)((N * 9 + 15) / 16), 10);
  dd_kernel<64, 2><<<g1, 32, 0, stream>>>(td0, W1b, W2b, bufA, bufB, N);
  tp_kernel<false><<<N, 64, 0, stream>>>(bufA, bufB, wpb, td1, hdr, grp, ent);

  // species embedding transform + fused output
  te_kernel<<<N / 16, 32, 0, stream>>>(Zs, emb, Wet, bet, te, N);
  final_kernel<<<(unsigned)(((size_t)N * 2 * NUM_LM * (FE_DIM / 4) + 255) / 256), 256, 0, stream>>>(
      y0, td0, td1, te, wf, out, N);
}